// PowerIteration_32220844655189
// MI455X (gfx1250) — compile-verified
//
#include <hip/hip_runtime.h>
#include <hip/hip_bf16.h>
#include <math.h>

// Problem constants (match reference setup_inputs)
#define BATCH    64
#define MN       2048
#define MAX_ITER 50
#define STOP_THRESH 2e-07f

typedef __attribute__((ext_vector_type(2))) float v2f;
typedef __attribute__((ext_vector_type(8))) float v8f;

// ---------------------------------------------------------------------------
// Init: v = v0, vlast = v0, flags = {done=0, done_old=0}
// ---------------------------------------------------------------------------
__global__ __launch_bounds__(256) void pi_init(const float* __restrict__ v0,
                                               float* __restrict__ v,
                                               float* __restrict__ vlast,
                                               int* __restrict__ flags) {
    int i = blockIdx.x * 256 + threadIdx.x;
    float x = v0[i];
    v[i] = x;
    vlast[i] = x;
    if (i == 0) { flags[0] = 0; flags[1] = 0; }
}

// ---------------------------------------------------------------------------
// Batched GEMV via V_WMMA_F32_16X16X4_F32.
// Block = 128 threads (4 waves). Each wave computes 16 rows of one batch.
// grid = (MN/64, BATCH). v for the batch staged in LDS (8KB) via the CDNA5
// async global->LDS copy path (ASYNCcnt), then broadcast as the B matrix.
// ---------------------------------------------------------------------------
__global__ __launch_bounds__(128) void matvec_wmma(const float* __restrict__ M,
                                                   const float* __restrict__ v,
                                                   float* __restrict__ vn) {
    __shared__ float vs[MN];
    const int b = blockIdx.y;
    const int t = threadIdx.x;

    // Async staging of v[b,:] into LDS: 128 threads x 16B x 4 = 8KB.
    // GLOBAL_LOAD_ASYNC_TO_LDS_B128: memory -> LDS with no VGPR round-trip,
    // tracked by ASYNCcnt. Each wave drains its own counter; __syncthreads
    // then publishes LDS to all 4 waves.
    {
        const float* gsrc = v + (size_t)b * MN + t * 4;
        unsigned lbase = (unsigned)(uintptr_t)(&vs[0]) + t * 16;
#pragma unroll
        for (int j = 0; j < 4; ++j) {
            unsigned laddr = lbase + j * 2048;                       // bytes
            unsigned long long gaddr = (unsigned long long)(uintptr_t)(gsrc + j * 512);
            asm volatile("global_load_async_to_lds_b128 %0, %1, off"
                         :: "v"(laddr), "v"(gaddr)
                         : "memory");
        }
        asm volatile("s_wait_asynccnt 0" ::: "memory");
    }
    __syncthreads();

    const int wave = t >> 5;
    const int lane = t & 31;
    const int row0 = blockIdx.x * 64 + wave * 16;
    const int rtile = lane & 15;          // row within 16-row tile
    const int khalf = (lane >> 4) << 1;   // K-pair: {0,1} or {2,3}

    const float* __restrict__ Mrow =
        M + ((size_t)b * MN + (size_t)(row0 + rtile)) * MN + khalf;

    v8f acc0 = {}, acc1 = {}, acc2 = {}, acc3 = {};

    auto step = [&](int k) {
        v2f a0 = *(const v2f*)(Mrow + k +  0);
        v2f a1 = *(const v2f*)(Mrow + k +  4);
        v2f a2 = *(const v2f*)(Mrow + k +  8);
        v2f a3 = *(const v2f*)(Mrow + k + 12);

        v2f b0 = { vs[k +  0 + khalf], vs[k +  1 + khalf] };
        v2f b1 = { vs[k +  4 + khalf], vs[k +  5 + khalf] };
        v2f b2 = { vs[k +  8 + khalf], vs[k +  9 + khalf] };
        v2f b3 = { vs[k + 12 + khalf], vs[k + 13 + khalf] };

        acc0 = __builtin_amdgcn_wmma_f32_16x16x4_f32(false, a0, false, b0,
                                                     (short)0, acc0, false, false);
        acc1 = __builtin_amdgcn_wmma_f32_16x16x4_f32(false, a1, false, b1,
                                                     (short)0, acc1, false, false);
        acc2 = __builtin_amdgcn_wmma_f32_16x16x4_f32(false, a2, false, b2,
                                                     (short)0, acc2, false, false);
        acc3 = __builtin_amdgcn_wmma_f32_16x16x4_f32(false, a3, false, b3,
                                                     (short)0, acc3, false, false);
    };

    // Main body: unconditional prefetch 512 floats (2KB) ahead; branch-free.
    int k = 0;
#pragma unroll 4
    for (; k < MN - 512; k += 16) {
        __builtin_prefetch(Mrow + k + 512, 0, 3);
        step(k);
    }
    // Tail: last 512 floats, no prefetch (avoids OOB and in-loop guards).
#pragma unroll 4
    for (; k < MN; k += 16) {
        step(k);
    }

    v8f acc = acc0 + acc1 + acc2 + acc3;  // elementwise == matrix add (same layout)

    // C/D layout: lane 0 VGPR r = C[r][0], lane 16 VGPR r = C[8+r][0].
    // All columns identical (B replicated), so lanes 0 and 16 hold the result.
    if (rtile == 0) {
        float* out = vn + (size_t)b * MN + row0 + ((lane >> 4) << 3);
#pragma unroll
        for (int r = 0; r < 8; ++r) out[r] = acc[r];
    }
}

// ---------------------------------------------------------------------------
// Per-batch: n = ||vn||_2 ; vn /= n (in place) ; partial[b] = sum (vn-vlast)^2
// One block of 256 threads per batch (2048 = 256*8 elements).
// ---------------------------------------------------------------------------
__global__ __launch_bounds__(256) void normalize_diff(float* __restrict__ vn,
                                                      const float* __restrict__ vlast,
                                                      float* __restrict__ partial) {
    __shared__ float red[256];
    const int b = blockIdx.x;
    const int t = threadIdx.x;
    const size_t base = (size_t)b * MN;

    float x[8];
    float s = 0.0f;
#pragma unroll
    for (int j = 0; j < 8; ++j) {
        x[j] = vn[base + t + j * 256];
        s += x[j] * x[j];
    }
    red[t] = s;
    __syncthreads();
    for (int off = 128; off > 0; off >>= 1) {
        if (t < off) red[t] += red[t + off];
        __syncthreads();
    }
    const float invn = 1.0f / sqrtf(red[0]);
    __syncthreads();  // everyone has read red[0] before reuse

    float d = 0.0f;
#pragma unroll
    for (int j = 0; j < 8; ++j) {
        float y = x[j] * invn;
        vn[base + t + j * 256] = y;
        float df = y - vlast[base + t + j * 256];
        d += df * df;
    }
    red[t] = d;
    __syncthreads();
    for (int off = 128; off > 0; off >>= 1) {
        if (t < off) red[t] += red[t + off];
        __syncthreads();
    }
    if (t == 0) partial[b] = red[0];
}

// ---------------------------------------------------------------------------
// Single block: global diff = sqrt(sum partial); flags[1]=old done;
// flags[0] = done | (diff < thresh)   (sticky, like the reference)
// ---------------------------------------------------------------------------
__global__ __launch_bounds__(64) void finalize_diff(const float* __restrict__ partial,
                                                    int* __restrict__ flags) {
    __shared__ float red[64];
    const int t = threadIdx.x;
    red[t] = partial[t];
    __syncthreads();
    for (int off = 32; off > 0; off >>= 1) {
        if (t < off) red[t] += red[t + off];
        __syncthreads();
    }
    if (t == 0) {
        int done = flags[0];
        flags[1] = done;  // done_old used to gate this iteration's commit
        float diff = sqrtf(red[0]);
        flags[0] = done | (diff < STOP_THRESH ? 1 : 0);
    }
}

// ---------------------------------------------------------------------------
// Commit (gated on OLD done): v = vn ; vlast = vn
// ---------------------------------------------------------------------------
__global__ __launch_bounds__(256) void commit(const float* __restrict__ vn,
                                              float* __restrict__ v,
                                              float* __restrict__ vlast,
                                              const int* __restrict__ flags) {
    if (flags[1]) return;  // frozen after convergence
    int i = blockIdx.x * 256 + threadIdx.x;
    float y = vn[i];
    v[i] = y;
    vlast[i] = y;
}

__global__ __launch_bounds__(256) void copy_out(const float* __restrict__ v,
                                                float* __restrict__ out) {
    int i = blockIdx.x * 256 + threadIdx.x;
    out[i] = v[i];
}

// ---------------------------------------------------------------------------
extern "C" void kernel_launch(void* const* d_in, const int* in_sizes, int n_in,
                              void* d_out, int out_size, void* d_ws, size_t ws_size,
                              hipStream_t stream) {
    const float* M  = (const float*)d_in[0];   // [64, 2048, 2048] fp32
    const float* v0 = (const float*)d_in[1];   // [64, 2048, 1] fp32
    float* out = (float*)d_out;                // [64, 2048] fp32

    const int NV = BATCH * MN;                 // 131072
    float* ws      = (float*)d_ws;
    float* vn      = ws;                       // raw/normalized matvec result
    float* v       = ws + NV;                  // current vector
    float* vlast   = ws + 2 * NV;              // previous vector
    float* partial = ws + 3 * NV;              // per-batch diff partials [64]
    int*   flags   = (int*)(ws + 3 * NV + BATCH); // [done, done_old]

    const int nblk = NV / 256;                 // 512

    pi_init<<<nblk, 256, 0, stream>>>(v0, v, vlast, flags);

    dim3 gmv(MN / 64, BATCH);                  // (32, 64)
    for (int it = 0; it < MAX_ITER; ++it) {
        matvec_wmma<<<gmv, 128, 0, stream>>>(M, v, vn);
        normalize_diff<<<BATCH, 256, 0, stream>>>(vn, vlast, partial);
        finalize_diff<<<1, 64, 0, stream>>>(partial, flags);
        commit<<<nblk, 256, 0, stream>>>(vn, v, vlast, flags);
    }

    copy_out<<<nblk, 256, 0, stream>>>(v, out);
}